// MyRNN_51608327029371
// MI455X (gfx1250) — compile-verified
//
#include <hip/hip_runtime.h>
#include <math.h>

// Elman RNN, L=2, C=1024, B=16, T=2048, fp32, on gfx1250 (wave32, WMMA, TDM).
#define C_DIM 1024
#define T_DIM 2048
#define B_DIM 16

typedef float v2f __attribute__((ext_vector_type(2)));
typedef float v8f __attribute__((ext_vector_type(8)));
typedef unsigned int v4u __attribute__((ext_vector_type(4)));
typedef int v8i __attribute__((ext_vector_type(8)));
typedef int v4i __attribute__((ext_vector_type(4)));

// TDM pads LDS dest by 2 DWORDs every 256 DWORDs -> LDS row stride 1032 floats.
// 1032 % 64 banks == 8 -> at most 2-way conflict on f32x2 fragment reads.
#define LDH 1032

#if __has_builtin(__builtin_amdgcn_tensor_load_to_lds)
#define HAS_TDM 1
#else
#define HAS_TDM 0
#endif

// ---------------------------------------------------------------------------
// Stage a rows x 1024 f32 tile (global row stride 1024) into LDS with padded
// stride LDH, via the Tensor Data Mover (D# per CDNA5 ISA ch.8).
// ---------------------------------------------------------------------------
#if HAS_TDM
__device__ __forceinline__ void tdm_load_2d(const float* gsrc, float* ldst, int rows) {
  unsigned lds = (unsigned)(size_t)ldst;                 // flat LDS addr: low 32b = LDS offset
  unsigned long long ga = (unsigned long long)(size_t)gsrc;
  // Group0: count=1 | lds_addr | global_addr[56:0] | type=2
  v4u g0 = { 1u, lds, (unsigned)ga, (unsigned)((ga >> 32) & 0x1FFFFFFu) | (2u << 30) };
  // Group1: data_size=4B, pad_enable, pad_interval=7 (256 dw), pad_amount=1 (2 dw),
  //         tensor_dim0=1024, tensor_dim1=rows, tile_dim0=1024, tile_dim1=rows,
  //         tensor_dim0_stride=1024
  v8i g1 = { (int)((2u << 16) | (1u << 20) | (7u << 22) | (1u << 25)),
             (int)(1024u << 16),
             (int)((unsigned)rows << 16),
             (int)(1024u << 16),
             rows, 1024, 0, 0 };
  v4i gz4 = { 0, 0, 0, 0 };
  v8i gz8 = { 0, 0, 0, 0, 0, 0, 0, 0 };
  // 6-arg toolchain variant: groups 0..3, extra group, cache policy.
  __builtin_amdgcn_tensor_load_to_lds(g0, g1, gz4, gz4, gz8, 0);
}
#endif

// wave0 issues the DMA (or all threads cooperative-load as fallback); barrier on exit.
__device__ __forceinline__ void stage_tile(const float* gsrc, float* ldst, int rows, int tid) {
#if HAS_TDM
  if ((tid >> 5) == 0) {
    tdm_load_2d(gsrc, ldst, rows);
    __builtin_amdgcn_s_wait_tensorcnt(0);
  }
#else
  for (int flat = tid * 4; flat < rows * 1024; flat += 256 * 4) {
    int m = flat >> 10, k = flat & 1023;
    v2f v0 = *(const v2f*)(gsrc + flat);
    v2f v1 = *(const v2f*)(gsrc + flat + 2);
    float* d = ldst + m * LDH + k + 2 * (k >> 8);        // same padded layout
    *(v2f*)(d) = v0;
    *(v2f*)(d + 2) = v1;
  }
#endif
  __syncthreads();
}

// ---------------------------------------------------------------------------
// Kernel 1: out[row,:] = in[row,:] @ W^T + bias   (rows = B*T = 32768)
// Block = 256 thr (8 waves), 64-row x 1024-col tile; in-place safe.
// ---------------------------------------------------------------------------
__global__ __launch_bounds__(256) void gemm_ih(const float* in, const float* __restrict__ W,
                                               const float* __restrict__ bias, float* out) {
  __shared__ float Alds[64 * LDH];
  const int tid = threadIdx.x;
  const long rowbase = (long)blockIdx.x * 64;

  stage_tile(in + rowbase * C_DIM, Alds, 64, tid);

  const int lane = tid & 31, wave = tid >> 5;
  const int lan16 = lane & 15;
  const int nhalf = lane >> 4;   // lanes 16..31: K+2 (A/B frags), M+8 (C/D)
  const int koff = nhalf * 2;

  for (int nt = 0; nt < 8; ++nt) {
    const int col0 = wave * 128 + nt * 16;           // this wave's 16-col tile
    const float bv = bias[col0 + lan16];             // C/D: n = lane%16
    v8f acc0, acc1, acc2, acc3;
    for (int r = 0; r < 8; ++r) { acc0[r] = bv; acc1[r] = bv; acc2[r] = bv; acc3[r] = bv; }

    const float* wrow = W + (long)(col0 + lan16) * C_DIM + koff;  // B[k,n] = W[n,k]
    for (int kb = 0; kb < 4; ++kb) {
      const int kg = kb * 256;                 // global K base
      const int kl = kg + kb * 2 + koff;       // LDS K base incl. TDM pad
      __builtin_prefetch(wrow + kg + 256, 0, 0);      // global_prefetch_b8
      for (int k = 0; k < 256; k += 4) {
        v2f bfrag = *(const v2f*)(wrow + kg + k);
        v2f a0 = *(const v2f*)(&Alds[( 0 + lan16) * LDH + kl + k]);
        v2f a1 = *(const v2f*)(&Alds[(16 + lan16) * LDH + kl + k]);
        v2f a2 = *(const v2f*)(&Alds[(32 + lan16) * LDH + kl + k]);
        v2f a3 = *(const v2f*)(&Alds[(48 + lan16) * LDH + kl + k]);
        acc0 = __builtin_amdgcn_wmma_f32_16x16x4_f32(false, a0, false, bfrag, (short)0, acc0, false, false);
        acc1 = __builtin_amdgcn_wmma_f32_16x16x4_f32(false, a1, false, bfrag, (short)0, acc1, false, false);
        acc2 = __builtin_amdgcn_wmma_f32_16x16x4_f32(false, a2, false, bfrag, (short)0, acc2, false, false);
        acc3 = __builtin_amdgcn_wmma_f32_16x16x4_f32(false, a3, false, bfrag, (short)0, acc3, false, false);
      }
    }
    for (int r = 0; r < 8; ++r) {
      int m = r + 8 * nhalf;
      long c = col0 + lan16;
      out[(rowbase +  0 + m) * (long)C_DIM + c] = acc0[r];
      out[(rowbase + 16 + m) * (long)C_DIM + c] = acc1[r];
      out[(rowbase + 32 + m) * (long)C_DIM + c] = acc2[r];
      out[(rowbase + 48 + m) * (long)C_DIM + c] = acc3[r];
    }
  }
}

// ---------------------------------------------------------------------------
// Kernel 2: persistent recurrence for one layer.
// 32 WGs x 256 thr; WG g owns output cols [32g,32g+32). W_hh slice (129KB) in
// LDS for all T steps (TDM-preloaded). Per step: TDM-stage h, WMMA chains
// (4-way K-split), LDS reduce, tanh, write h slice, epoch barrier.
// ---------------------------------------------------------------------------
__global__ __launch_bounds__(256) void recur(const float* __restrict__ W,
                                             const float* __restrict__ bias,
                                             float* ZY, float* __restrict__ hT,
                                             float* hbuf, unsigned* cnt) {
  __shared__ float Wl[32 * LDH];
  __shared__ float hl[B_DIM * LDH];
  __shared__ float red[6 * 256];
  __shared__ float bl[32];

  const int tid = threadIdx.x;
  const int colbase = blockIdx.x * 32;

  if (tid < 32) bl[tid] = bias[colbase + tid];
  stage_tile(W + (long)colbase * C_DIM, Wl, 32, tid);   // one-time W_hh slice preload

  const int lane = tid & 31, wave = tid >> 5;
  const int lan16 = lane & 15, nhalf = lane >> 4, koff = nhalf * 2;
  const int nt = wave & 1;        // which 16-col tile of this WG's 32 cols
  const int kq = wave >> 1;       // K-quarter 0..3
  const int n_local = nt * 16 + lan16;
  const int pbase = kq * 2 + koff;  // TDM pad offset: constant per wave

  for (int t = 0; t < T_DIM; ++t) {
    const int cur = t & 1, nxt = cur ^ 1;

    // Stage full h_{t-1} [16x1024] from global exchange buffer into LDS (TDM)
    stage_tile(hbuf + cur * (B_DIM * C_DIM), hl, B_DIM, tid);

    // Partial [16x16] tile for (nt, kq): 64 chained f32 WMMAs
    v8f acc;
    for (int r = 0; r < 8; ++r) acc[r] = 0.0f;
    const int k0 = kq * 256;
    for (int k = k0; k < k0 + 256; k += 4) {
      v2f a = *(const v2f*)(&hl[lan16 * LDH + k + pbase]);
      v2f b = *(const v2f*)(&Wl[n_local * LDH + k + pbase]);
      acc = __builtin_amdgcn_wmma_f32_16x16x4_f32(false, a, false, b, (short)0, acc, false, false);
    }
    if (kq > 0) {
      float* rp = red + ((kq - 1) * 2 + nt) * 256;
      for (int r = 0; r < 8; ++r) rp[r * 32 + lane] = acc[r];
    }
    __syncthreads();

    if (kq == 0) {  // reduce partials, add Z + bias, tanh, write out + h-exchange
      for (int q = 1; q < 4; ++q) {
        const float* rp = red + ((q - 1) * 2 + nt) * 256;
        for (int r = 0; r < 8; ++r) acc[r] += rp[r * 32 + lane];
      }
      const float bv = bl[n_local];
      const long col = colbase + n_local;
      for (int r = 0; r < 8; ++r) {
        int m = r + 8 * nhalf;                       // batch index
        long idx = (long)m * T_DIM * C_DIM + (long)t * C_DIM + col;
        float h = tanhf(ZY[idx] + bv + acc[r]);
        ZY[idx] = h;
        hbuf[nxt * (B_DIM * C_DIM) + m * C_DIM + col] = h;
        if (t == T_DIM - 1) hT[m * C_DIM + col] = h;
      }
      __threadfence();  // make h slice visible device-wide before signaling
    }
    __syncthreads();

    // Epoch barrier across the 32 co-resident WGs
    if (tid == 0) {
      __hip_atomic_fetch_add(cnt, 1u, __ATOMIC_ACQ_REL, __HIP_MEMORY_SCOPE_AGENT);
      const unsigned target = 32u * (unsigned)(t + 1);
      while (__hip_atomic_load(cnt, __ATOMIC_ACQUIRE, __HIP_MEMORY_SCOPE_AGENT) < target) {
        __builtin_amdgcn_s_sleep(2);
      }
    }
    __syncthreads();
  }
}

// Init per-layer state: barrier counter = 0, h exchange buf[0] = h_0[layer].
__global__ void init_ws(const float* __restrict__ h0, float* hbuf, unsigned* cnt) {
  int i = blockIdx.x * 256 + threadIdx.x;
  if (i < B_DIM * C_DIM) hbuf[i] = h0[i];
  if (i == 0) *cnt = 0u;
}

extern "C" void kernel_launch(void* const* d_in, const int* in_sizes, int n_in,
                              void* d_out, int out_size, void* d_ws, size_t ws_size,
                              hipStream_t stream) {
  const float* X    = (const float*)d_in[0];
  const float* h_0  = (const float*)d_in[1];
  const float* W_ih = (const float*)d_in[2];
  const float* b_ih = (const float*)d_in[3];
  const float* W_hh = (const float*)d_in[4];
  const float* b_hh = (const float*)d_in[5];

  float* ret = (float*)d_out;                              // [B,T,C]
  float* hT  = ret + (long)B_DIM * T_DIM * C_DIM;          // [L,B,C]
  unsigned* cnt = (unsigned*)d_ws;
  float* hbuf   = (float*)((char*)d_ws + 256);             // [2][B*C]

  const int rows = B_DIM * T_DIM;                          // 32768
  const long WW = (long)C_DIM * C_DIM;

  // Layer 0: Z0 = X @ W_ih0^T + b0 -> ret; recurrence turns ret into Y0
  gemm_ih<<<rows / 64, 256, 0, stream>>>(X, W_ih, b_ih, ret);
  init_ws<<<(B_DIM * C_DIM + 255) / 256, 256, 0, stream>>>(h_0, hbuf, cnt);
  recur<<<32, 256, 0, stream>>>(W_hh, b_hh, ret, hT, hbuf, cnt);

  // Layer 1: Z1 = Y0 @ W_ih1^T + b1 (in place); recurrence turns ret into ys
  gemm_ih<<<rows / 64, 256, 0, stream>>>(ret, W_ih + WW, b_ih + C_DIM, ret);
  init_ws<<<(B_DIM * C_DIM + 255) / 256, 256, 0, stream>>>(h_0 + B_DIM * C_DIM, hbuf, cnt);
  recur<<<32, 256, 0, stream>>>(W_hh + WW, b_hh + C_DIM, ret, hT + B_DIM * C_DIM, hbuf, cnt);
}